// TripletRankingLoss_12360915878278
// MI455X (gfx1250) — compile-verified
//
#include <hip/hip_runtime.h>

typedef __attribute__((ext_vector_type(2))) float v2f;
typedef __attribute__((ext_vector_type(8))) float v8f;

#define N_ROWS 16384
#define E_DIM  1024
#define MARGIN_F 1.0f

__global__ void TRL_zero_ws(float* ws) {
    if (threadIdx.x < 4) ws[threadIdx.x] = 0.0f;
}

__device__ __forceinline__ float diag_pick(const v8f& v, int r) {
    float d = v[0];
    if (r == 1) d = v[1];
    if (r == 2) d = v[2];
    if (r == 3) d = v[3];
    if (r == 4) d = v[4];
    if (r == 5) d = v[5];
    if (r == 6) d = v[6];
    if (r == 7) d = v[7];
    return d;
}

// One wave (32 lanes) owns a 16-row tile. Three f32 WMMA accumulators:
//   accAnchor = audio_tile   x query_tile^T
//   accAimp   = audio_gather x query_tile^T
//   accQimp   = audio_tile   x query_gather^T
// Diagonals give the per-row dot products exactly (f32 WMMA, no precision loss).
__global__ __launch_bounds__(256) void TRL_wmma_kernel(
    const float* __restrict__ audio, const float* __restrict__ query,
    const int* __restrict__ a_neg, const int* __restrict__ q_neg,
    float* __restrict__ ws)
{
    const int lane = threadIdx.x & 31;
    const int wave = threadIdx.x >> 5;
    const int tile = blockIdx.x * 8 + wave;     // 1024 tiles of 16 rows

    const int m  = lane & 15;                   // row within tile this lane feeds
    const int hi = lane >> 4;                   // K sub-pair select (0 or 1)

    const int row   = tile * 16 + m;
    const int rowAn = a_neg[row];               // imposter audio row
    const int rowBn = q_neg[row];               // imposter query row

    // Each lane streams contiguous float2 pairs from its row:
    // chunk c covers columns [4c, 4c+3]; this lane takes cols 4c+2*hi, +1.
    const v2f* pA  = (const v2f*)(audio + (size_t)row   * E_DIM) + hi;
    const v2f* pAn = (const v2f*)(audio + (size_t)rowAn * E_DIM) + hi;
    const v2f* pB  = (const v2f*)(query + (size_t)row   * E_DIM) + hi;
    const v2f* pBn = (const v2f*)(query + (size_t)rowBn * E_DIM) + hi;

    v8f accAnchor = {};
    v8f accAimp   = {};
    v8f accQimp   = {};

    #pragma unroll 4
    for (int c = 0; c < E_DIM / 4; ++c) {
        v2f a  = pA [2 * c];
        v2f an = pAn[2 * c];
        v2f b  = pB [2 * c];
        v2f bn = pBn[2 * c];
        accAnchor = __builtin_amdgcn_wmma_f32_16x16x4_f32(
            false, a,  false, b,  (short)0, accAnchor, false, false);
        accAimp   = __builtin_amdgcn_wmma_f32_16x16x4_f32(
            false, an, false, b,  (short)0, accAimp,   false, false);
        accQimp   = __builtin_amdgcn_wmma_f32_16x16x4_f32(
            false, a,  false, bn, (short)0, accQimp,   false, false);
    }

    // Diagonal extraction from 16x16 f32 C/D layout:
    //   M<8  -> lane M,     VGPR M
    //   M>=8 -> lane M+16,  VGPR M-8   (i.e. lanes 24..31)
    int r = -1;
    if (lane < 8)        r = lane;
    else if (lane >= 24) r = lane - 24;

    float anchor = 0.f, aimp = 0.f, qimp = 0.f, hinge = 0.f;
    if (r >= 0) {
        anchor = diag_pick(accAnchor, r);
        aimp   = diag_pick(accAimp,   r);
        qimp   = diag_pick(accQimp,   r);
        float a2q = MARGIN_F + qimp - anchor; a2q = a2q > 0.f ? a2q : 0.f;
        float q2a = MARGIN_F + aimp - anchor; q2a = q2a > 0.f ? q2a : 0.f;
        hinge = a2q + q2a;
    }

    // Wave32 butterfly reduction; inactive lanes contribute zero.
    #pragma unroll
    for (int off = 16; off > 0; off >>= 1) {
        hinge  += __shfl_xor(hinge,  off, 32);
        anchor += __shfl_xor(anchor, off, 32);
        aimp   += __shfl_xor(aimp,   off, 32);
        qimp   += __shfl_xor(qimp,   off, 32);
    }

    if (lane == 0) {
        atomicAdd(&ws[0], hinge);
        atomicAdd(&ws[1], anchor);
        atomicAdd(&ws[2], aimp);
        atomicAdd(&ws[3], qimp);
    }
}

__global__ void TRL_finalize(const float* __restrict__ ws, float* __restrict__ out) {
    if (threadIdx.x == 0) {
        const float invN = 1.0f / (float)N_ROWS;
        out[0] = ws[0] * invN;   // loss
        out[1] = ws[1] * invN;   // mean(anchor)
        out[2] = ws[2] * invN;   // mean(a_imp)
        out[3] = ws[3] * invN;   // mean(q_imp)
    }
}

extern "C" void kernel_launch(void* const* d_in, const int* in_sizes, int n_in,
                              void* d_out, int out_size, void* d_ws, size_t ws_size,
                              hipStream_t stream) {
    const float* audio = (const float*)d_in[0];
    const float* query = (const float*)d_in[1];
    const int*   a_neg = (const int*)d_in[2];
    const int*   q_neg = (const int*)d_in[3];
    float* ws  = (float*)d_ws;
    float* out = (float*)d_out;

    TRL_zero_ws<<<1, 32, 0, stream>>>(ws);

    // 1024 tiles of 16 rows; 8 waves per 256-thread block -> 128 blocks.
    TRL_wmma_kernel<<<(N_ROWS / 16) / 8, 256, 0, stream>>>(audio, query, a_neg, q_neg, ws);

    TRL_finalize<<<1, 32, 0, stream>>>(ws, out);
}